// SparseAttention_50964081934577
// MI455X (gfx1250) — compile-verified
//
#include <hip/hip_runtime.h>
#include <hip/hip_bf16.h>

// ---------------------------------------------------------------------------
// SparseAttention for MI455X (gfx1250), fp32 end-to-end via V_WMMA_F32_16X16X4_F32.
// B=2, S=2048, H=1024, nH=16, dH=64, local band +-32, stride 8.
// Round 2: batched fragment loads per k-chunk so global_load_b64 groups overlap
// the WMMA blocks (kills the s_wait_loadcnt-before-every-wmma pattern), and a
// larger 32x64 wave tile in the GEMM for better load:wmma ratio.
// ---------------------------------------------------------------------------

typedef __attribute__((ext_vector_type(2))) float v2f;
typedef __attribute__((ext_vector_type(8))) float v8f;

#define S_LEN   2048
#define HID     1024
#define NHEADS  16
#define DHEAD   64
#define LSTR    2052                      // 2048 + 4 pad: kills LDS bank conflicts
#define SMEM_FLOATS (16 * LSTR + 1024)    // P strip + split-K reduction scratch

__device__ __forceinline__ v8f wmma_f32(v2f a, v2f b, v8f c) {
    // D(16x16,f32) = A(16x4,f32) * B(4x16,f32) + C
    return __builtin_amdgcn_wmma_f32_16x16x4_f32(
        /*neg_a=*/false, a, /*neg_b=*/false, b,
        /*c_mod=*/(short)0, c, /*reuse_a=*/false, /*reuse_b=*/false);
}

// ---------------------------------------------------------------------------
// Y[m,n] = sum_k X[m,k] * W[n,k] + bias[n]   (nn.Linear: y = x @ W.T + b)
// M = 4096, N = K = 1024.  Block = 4 waves; block tile 128(M) x 64(N);
// each wave: 32(M) x 64(N), 8 accumulators.  K-chunk of 16: batch-load
// 8 A-frags + 16 B-frags (24 global_load_b64 in flight), then 32 WMMAs.
// head_layout=1 scatters Y into [B, nH, S, dH] for the attention kernel.
// ---------------------------------------------------------------------------
__global__ __launch_bounds__(128) void gemm_bias_kernel(
    const float* __restrict__ X, const float* __restrict__ W,
    const float* __restrict__ bias, float* __restrict__ Y,
    int head_layout)
{
    const int K = HID;
    const int wave = threadIdx.x >> 5;
    const int lane = threadIdx.x & 31;
    const int lm = lane & 15;          // row/col within 16-tile
    const int kh = lane >> 4;          // k-half selector (ISA f32 A/B layout)

    const int m0 = blockIdx.x * 128 + wave * 32;
    const int n0 = blockIdx.y * 64;

    const float* xrow0 = X + (long)(m0 + lm) * K + 2 * kh;
    const float* xrow1 = X + (long)(m0 + 16 + lm) * K + 2 * kh;
    const float* wrow[4];
#pragma unroll
    for (int t = 0; t < 4; ++t)
        wrow[t] = W + (long)(n0 + t * 16 + lm) * K + 2 * kh;

    v8f acc[2][4];
#pragma unroll
    for (int mh = 0; mh < 2; ++mh)
#pragma unroll
        for (int t = 0; t < 4; ++t) acc[mh][t] = v8f{};

#pragma unroll 1
    for (int k0 = 0; k0 < K; k0 += 16) {
        v2f a[2][4];     // [m-half][substep]
        v2f b[4][4];     // [n-tile][substep]
#pragma unroll
        for (int s = 0; s < 4; ++s) {
            a[0][s] = *(const v2f*)(xrow0 + k0 + 4 * s);
            a[1][s] = *(const v2f*)(xrow1 + k0 + 4 * s);
        }
#pragma unroll
        for (int t = 0; t < 4; ++t)
#pragma unroll
            for (int s = 0; s < 4; ++s)
                b[t][s] = *(const v2f*)(wrow[t] + k0 + 4 * s);

#pragma unroll
        for (int s = 0; s < 4; ++s)
#pragma unroll
            for (int t = 0; t < 4; ++t) {
                acc[0][t] = wmma_f32(a[0][s], b[t][s], acc[0][t]);
                acc[1][t] = wmma_f32(a[1][s], b[t][s], acc[1][t]);
            }
    }

#pragma unroll
    for (int mh = 0; mh < 2; ++mh)
#pragma unroll
        for (int t = 0; t < 4; ++t) {
            const int n = n0 + t * 16 + lm;
            const float bv = bias[n];
#pragma unroll
            for (int r = 0; r < 8; ++r) {
                const int m = m0 + mh * 16 + kh * 8 + r;   // D: M=(L>>4)*8+r
                const float val = acc[mh][t][r] + bv;
                if (head_layout) {
                    const int bb = m >> 11;                // / S_LEN
                    const int s  = m & (S_LEN - 1);
                    const int h  = n >> 6;                 // / DHEAD
                    const int d  = n & (DHEAD - 1);
                    Y[(((long)bb * NHEADS + h) * S_LEN + s) * DHEAD + d] = val;
                } else {
                    Y[(long)m * HID + n] = val;
                }
            }
        }
}

// ---------------------------------------------------------------------------
// One block per (b, h, 16-row q tile).  8 waves, 135 KB dynamic LDS.
//  Phase 1: S = Q K^T * 0.125, masked, into LDS P[16][2052]
//           (batch: 16 K-fragments loaded per column tile, then 16 WMMAs)
//  Phase 2: row softmax (wave32 shfl_xor reductions), normalize in LDS,
//           stream dense attn rows out as float4 (masked cols are exactly 0
//           after exp(-1e9 - max) underflow, matching the reference).
//  Phase 3: context = P @ V via WMMA, 64-deep chunks (16 LDS A-frags +
//           32 V words batched, then 16 WMMAs), split-K x2 + LDS reduction.
// ---------------------------------------------------------------------------
__global__ __launch_bounds__(256) void attn_kernel(
    const float* __restrict__ Q, const float* __restrict__ Kmat,
    const float* __restrict__ V, float* __restrict__ attn,
    float* __restrict__ ctx)
{
    extern __shared__ float smem[];
    float* P   = smem;                 // 16 * LSTR
    float* red = smem + 16 * LSTR;     // 4 tiles * 256 floats

    const int wave = threadIdx.x >> 5;
    const int lane = threadIdx.x & 31;
    const int lm = lane & 15;
    const int kh = lane >> 4;

    const int qt = blockIdx.x, h = blockIdx.y, bb = blockIdx.z;
    const int q0 = qt * 16;

    const float* Qh = Q    + ((long)(bb * NHEADS + h) * S_LEN) * DHEAD;
    const float* Kh = Kmat + ((long)(bb * NHEADS + h) * S_LEN) * DHEAD;
    const float* Vh = V    + ((long)(bb * NHEADS + h) * S_LEN) * DHEAD;

    // A fragments for the whole 64-deep Q tile (16 k-steps of 4)
    v2f qa[16];
    {
        const float* qrow = Qh + (long)(q0 + lm) * DHEAD + 2 * kh;
#pragma unroll
        for (int s = 0; s < 16; ++s) qa[s] = *(const v2f*)(qrow + 4 * s);
    }

    // ---- Phase 1: masked scores ----
    const float scale = 0.125f;   // 1/sqrt(64)
#pragma unroll 1
    for (int i = 0; i < 16; ++i) {
        const int ct = wave + 8 * i;           // 128 column tiles / 8 waves
        const int n0 = ct * 16;
        const float* krow = Kh + (long)(n0 + lm) * DHEAD + 2 * kh;

        v2f b[16];                             // batch: 16 loads in flight
#pragma unroll
        for (int s = 0; s < 16; ++s) b[s] = *(const v2f*)(krow + 4 * s);

        v8f acc = {};
#pragma unroll
        for (int s = 0; s < 16; ++s) acc = wmma_f32(qa[s], b[s], acc);

        const int j = n0 + lm;
#pragma unroll
        for (int r = 0; r < 8; ++r) {
            const int lrow = kh * 8 + r;
            const int ii = q0 + lrow;
            const bool keep = ((unsigned)(j - ii + 32) < 64u) || ((j & 7) == 0);
            P[lrow * LSTR + j] = keep ? acc[r] * scale : -1.0e9f;
        }
    }
    __syncthreads();

    // ---- Phase 2: softmax + dense attn store ----
    float* attn_base = attn + ((long)(bb * NHEADS + h) * S_LEN + q0) * S_LEN;
#pragma unroll
    for (int rr = 0; rr < 2; ++rr) {
        const int row = wave + 8 * rr;         // 16 rows / 8 waves
        float* prow = P + row * LSTR;

        float mx = -3.0e38f;
#pragma unroll 4
        for (int t = 0; t < 16; ++t) {
            float4 x = *(const float4*)(prow + (lane + 32 * t) * 4);
            mx = fmaxf(mx, fmaxf(fmaxf(x.x, x.y), fmaxf(x.z, x.w)));
        }
#pragma unroll
        for (int off = 16; off >= 1; off >>= 1)
            mx = fmaxf(mx, __shfl_xor(mx, off, 32));

        float sum = 0.0f;
#pragma unroll 4
        for (int t = 0; t < 16; ++t) {
            float4 x = *(const float4*)(prow + (lane + 32 * t) * 4);
            x.x = __expf(x.x - mx); x.y = __expf(x.y - mx);
            x.z = __expf(x.z - mx); x.w = __expf(x.w - mx);
            sum += x.x + x.y + x.z + x.w;
            *(float4*)(prow + (lane + 32 * t) * 4) = x;
        }
#pragma unroll
        for (int off = 16; off >= 1; off >>= 1)
            sum += __shfl_xor(sum, off, 32);
        const float inv = 1.0f / sum;

        float* arow = attn_base + (long)row * S_LEN;
#pragma unroll 4
        for (int t = 0; t < 16; ++t) {
            const int c = (lane + 32 * t) * 4;
            float4 x = *(const float4*)(prow + c);
            x.x *= inv; x.y *= inv; x.z *= inv; x.w *= inv;
            *(float4*)(prow + c) = x;        // normalized P for phase 3
            *(float4*)(arow + c) = x;        // coalesced 537 MB attn stream
        }
    }
    __syncthreads();

    // ---- Phase 3: context = P @ V (split-K over 2 wave groups) ----
    const int t  = wave & 3;     // n-tile: 4 x 16 = 64 head dims
    const int hk = wave >> 2;    // k-half: [0,1024) or [1024,2048)
    const int n0 = t * 16;
    v8f acc = {};
    const float* prowA = P + (long)lm * LSTR + 2 * kh;   // A[m][k] = P[m][k]
    const int kbeg = hk * 1024;
#pragma unroll 1
    for (int k0 = kbeg; k0 < kbeg + 1024; k0 += 64) {
        v2f a[16];
        v2f b[16];
#pragma unroll
        for (int s = 0; s < 16; ++s)
            a[s] = *(const v2f*)(prowA + k0 + 4 * s);     // ds_load_b64
#pragma unroll
        for (int s = 0; s < 16; ++s) {
            const float* vp = Vh + (long)(k0 + 4 * s + 2 * kh) * DHEAD + n0 + lm;
            b[s].x = vp[0];                                // B[k][n] = V[k][n]
            b[s].y = vp[DHEAD];
        }
#pragma unroll
        for (int s = 0; s < 16; ++s) acc = wmma_f32(a[s], b[s], acc);
    }
    if (hk == 1) {
#pragma unroll
        for (int r = 0; r < 8; ++r) red[t * 256 + lane * 8 + r] = acc[r];
    }
    __syncthreads();
    if (hk == 0) {
#pragma unroll
        for (int r = 0; r < 8; ++r) {
            const float val = acc[r] + red[t * 256 + lane * 8 + r];
            const int m = q0 + kh * 8 + r;
            ctx[((long)bb * S_LEN + m) * HID + h * DHEAD + n0 + lm] = val;
        }
    }
}

// ---------------------------------------------------------------------------
extern "C" void kernel_launch(void* const* d_in, const int* in_sizes, int n_in,
                              void* d_out, int out_size, void* d_ws, size_t ws_size,
                              hipStream_t stream) {
    const float* q  = (const float*)d_in[0];
    const float* k  = (const float*)d_in[1];
    const float* v  = (const float*)d_in[2];
    const float* Wq = (const float*)d_in[3];
    const float* bq = (const float*)d_in[4];
    const float* Wk = (const float*)d_in[5];
    const float* bk = (const float*)d_in[6];
    const float* Wv = (const float*)d_in[7];
    const float* bv = (const float*)d_in[8];
    const float* Wo = (const float*)d_in[9];
    const float* bo = (const float*)d_in[10];

    float* out  = (float*)d_out;                          // [2,2048,1024]
    float* attn = out + (size_t)2 * S_LEN * HID;          // [2,16,2048,2048]

    const size_t per = (size_t)2 * NHEADS * S_LEN * DHEAD; // 4,194,304 floats
    float* Qws = (float*)d_ws;
    float* Kws = Qws + per;
    float* Vws = Kws + per;
    float* Ctx = Vws + per;

    dim3 ggrid(32, 16), gblock(128);
    gemm_bias_kernel<<<ggrid, gblock, 0, stream>>>(q, Wq, bq, Qws, 1);
    gemm_bias_kernel<<<ggrid, gblock, 0, stream>>>(k, Wk, bk, Kws, 1);
    gemm_bias_kernel<<<ggrid, gblock, 0, stream>>>(v, Wv, bv, Vws, 1);

    const size_t smem_bytes = SMEM_FLOATS * sizeof(float);   // 135,424 B (<320 KB WGP LDS)
    hipFuncSetAttribute(reinterpret_cast<const void*>(attn_kernel),
                        hipFuncAttributeMaxDynamicSharedMemorySize, (int)smem_bytes);
    attn_kernel<<<dim3(S_LEN / 16, NHEADS, 2), dim3(256), smem_bytes, stream>>>(
        Qws, Kws, Vws, attn, Ctx);

    gemm_bias_kernel<<<ggrid, gblock, 0, stream>>>(Ctx, Wo, bo, out, 0);
}